// PropagLayer_57655640982055
// MI455X (gfx1250) — compile-verified
//
#include <hip/hip_runtime.h>

typedef float v2f __attribute__((ext_vector_type(2)));
typedef float v8f __attribute__((ext_vector_type(8)));

#define NCOEF 32  // Taylor degree about t=0.5; truncation error << fp32 eps

// ---------------------------------------------------------------------------
// Kernel 1: build Taylor coefficients b_j = sum_k Gamma_k e^{0.5 z_k} z_k^j/j!
// Single wave32. Reduction over the 128 modes done with V_WMMA_F32_16X16X4_F32
// (A = coefficient samples, B = ones -> column sum), accumulating over 32
// chunks of 4 modes. Per ISA A-layout (16x4 f32): lane L holds row M=L%16,
// lanes 0-15 supply K=0,1 (v0,v1), lanes 16-31 supply K=2,3.
// ---------------------------------------------------------------------------
__global__ __launch_bounds__(32) void propag_coef_kernel(
    const float* __restrict__ params,   // [3,128]: Gamma, omega, gamma
    float* __restrict__ coef)           // out: 32 complex as interleaved f32
{
  const int lane = threadIdx.x;   // 0..31
  const int jlo  = lane & 15;     // WMMA row (coefficient index within group)
  const int half = lane >> 4;     // 0 -> K=0,1 ; 1 -> K=2,3

  v2f ones; ones.x = 1.0f; ones.y = 1.0f;
  v8f accRL = {0.f,0.f,0.f,0.f,0.f,0.f,0.f,0.f};  // Re, j=0..15
  v8f accRH = accRL;                               // Re, j=16..31
  v8f accIL = accRL;                               // Im, j=0..15
  v8f accIH = accRL;                               // Im, j=16..31

  for (int c = 0; c < 32; ++c) {                   // 32 chunks of 4 modes
    float aRL[2], aRH[2], aIL[2], aIH[2];
#pragma unroll
    for (int m = 0; m < 2; ++m) {                  // the lane's two A VGPRs
      const int k = c * 4 + half * 2 + m;          // mode index (K position)
      const float G  =  params[k];                 // Gamma_k
      const float zi =  params[128 + k];           // Im z = omega_k
      const float zr = -params[256 + k];           // Re z = -gamma_k
      float sn, cs;
      __sincosf(0.5f * zi, &sn, &cs);
      const float e = G * __expf(0.5f * zr);       // Gamma * |e^{z/2}|
      float cre = e * cs;                          // c_0 = Gamma e^{0.5 z}
      float cim = e * sn;
      float rl = 0.f, rh = 0.f, il = 0.f, ih = 0.f;
#pragma unroll
      for (int j = 0; j < NCOEF; ++j) {            // ladder c_{j+1}=c_j*z/(j+1)
        if (j == jlo)      { rl = cre; il = cim; }
        if (j == jlo + 16) { rh = cre; ih = cim; }
        const float inv = 1.0f / (float)(j + 1);   // compile-time literal
        const float nre = (cre * zr - cim * zi) * inv;
        const float nim = (cre * zi + cim * zr) * inv;
        cre = nre; cim = nim;
      }
      aRL[m] = rl; aRH[m] = rh; aIL[m] = il; aIH[m] = ih;
    }
    v2f vRL, vRH, vIL, vIH;
    vRL.x = aRL[0]; vRL.y = aRL[1];
    vRH.x = aRH[0]; vRH.y = aRH[1];
    vIL.x = aIL[0]; vIL.y = aIL[1];
    vIH.x = aIH[0]; vIH.y = aIH[1];
    // D = A x ones + C : per-row sum of this chunk's 4 modes, accumulated.
    accRL = __builtin_amdgcn_wmma_f32_16x16x4_f32(false, vRL, false, ones, (short)0, accRL, false, false);
    accRH = __builtin_amdgcn_wmma_f32_16x16x4_f32(false, vRH, false, ones, (short)0, accRH, false, false);
    accIL = __builtin_amdgcn_wmma_f32_16x16x4_f32(false, vIL, false, ones, (short)0, accIL, false, false);
    accIH = __builtin_amdgcn_wmma_f32_16x16x4_f32(false, vIH, false, ones, (short)0, accIH, false, false);
  }

  // D layout: VGPR r -> M=r (lanes 0-15, N=lane) / M=r+8 (lanes 16-31).
  // Column N=0 (lanes 0 and 16) carries the full reduced vector.
  if (jlo == 0) {
    const int base = half * 8;  // lane 0: rows 0-7; lane 16: rows 8-15
#pragma unroll
    for (int r = 0; r < 8; ++r) {
      coef[2 * (base + r) + 0]      = accRL[r];
      coef[2 * (base + r) + 1]      = accIL[r];
      coef[2 * (base + r + 16) + 0] = accRH[r];
      coef[2 * (base + r + 16) + 1] = accIH[r];
    }
  }
}

// ---------------------------------------------------------------------------
// Kernel 2: evaluate f(t) = sum_j b_j (t-0.5)^j via complex Horner (real u).
// 4 points per thread; b128 load of t, two b128 stores of interleaved re/im.
// Memory-bound: ~12.6 MB total traffic.
// ---------------------------------------------------------------------------
__global__ __launch_bounds__(256) void propag_eval_kernel(
    const float* __restrict__ tg, const float* __restrict__ coef,
    float* __restrict__ out, int n)
{
  const int i0 = (blockIdx.x * 256 + (int)threadIdx.x) * 4;

  float cr[NCOEF], ci[NCOEF];
#pragma unroll
  for (int j = 0; j < NCOEF; ++j) { cr[j] = coef[2*j]; ci[j] = coef[2*j+1]; }

  if (i0 + 4 <= n) {
    const float4 t = *reinterpret_cast<const float4*>(tg + i0);
    float u[4] = { t.x - 0.5f, t.y - 0.5f, t.z - 0.5f, t.w - 0.5f };
    float re[4], im[4];
#pragma unroll
    for (int e = 0; e < 4; ++e) { re[e] = cr[NCOEF-1]; im[e] = ci[NCOEF-1]; }
#pragma unroll
    for (int j = NCOEF - 2; j >= 0; --j) {
#pragma unroll
      for (int e = 0; e < 4; ++e) {
        re[e] = fmaf(re[e], u[e], cr[j]);
        im[e] = fmaf(im[e], u[e], ci[j]);
      }
    }
    float4 o01, o23;
    o01.x = re[0]; o01.y = im[0]; o01.z = re[1]; o01.w = im[1];
    o23.x = re[2]; o23.y = im[2]; o23.z = re[3]; o23.w = im[3];
    reinterpret_cast<float4*>(out + 2 * i0)[0] = o01;
    reinterpret_cast<float4*>(out + 2 * i0)[1] = o23;
  } else {
    for (int i = i0; i < n; ++i) {
      const float u = tg[i] - 0.5f;
      float re = cr[NCOEF-1], im = ci[NCOEF-1];
#pragma unroll
      for (int j = NCOEF - 2; j >= 0; --j) {
        re = fmaf(re, u, cr[j]);
        im = fmaf(im, u, ci[j]);
      }
      out[2*i]   = re;
      out[2*i+1] = im;
    }
  }
}

extern "C" void kernel_launch(void* const* d_in, const int* in_sizes, int n_in,
                              void* d_out, int out_size, void* d_ws, size_t ws_size,
                              hipStream_t stream) {
  const float* params = (const float*)d_in[0];   // [3,128]
  const float* tg     = (const float*)d_in[1];   // [T]
  float* out  = (float*)d_out;                   // complex64[T] as 2T floats
  float* coef = (float*)d_ws;                    // 64 floats scratch
  const int T = in_sizes[1];

  propag_coef_kernel<<<1, 32, 0, stream>>>(params, coef);

  const int per_block = 256 * 4;
  const int blocks = (T + per_block - 1) / per_block;
  propag_eval_kernel<<<blocks, 256, 0, stream>>>(tg, coef, out, T);
}